// ManhattanAttention_39024072852183
// MI455X (gfx1250) — compile-verified
//
#include <hip/hip_runtime.h>
#include <stdint.h>

#define B_   2
#define SQ_  1024
#define SK_  1024
#define D_   64
#define DV_  64
#define NEG_ (-1e9f)

#define KT      32             // keys per LDS tile
#define VSTRIDE (DV_ + 4)      // padded V row stride (floats)
#define WAVES   2
#define BLOCK   (WAVES * 32)   // 64 threads
#define QPW     16             // queries per wave
#define QPB     (WAVES * QPW)  // 32 queries per block
#define NQB     (SQ_ / QPB)    // 32 query-blocks per batch
#define NT16    (SQ_ / 16)     // 64 16-row query tiles per batch
#define SPLIT   4              // K-dimension split (flash-decode style)

typedef float v2f __attribute__((ext_vector_type(2)));
typedef float v8f __attribute__((ext_vector_type(8)));

__device__ __forceinline__ void async_ld_b128(uint32_t lds_off, uint64_t gaddr) {
  asm volatile("global_load_async_to_lds_b128 %0, %1, off"
               :: "v"(lds_off), "v"(gaddr) : "memory");
}

// PARTIAL=true : process SK_/SPLIT keys, write unnormalized O + (m,l) to workspace
// PARTIAL=false: process all SK_ keys, write normalized output directly
template <bool PARTIAL>
__global__ __launch_bounds__(BLOCK) __attribute__((amdgpu_waves_per_eu(4)))
void mattn_main(const float* __restrict__ q, const float* __restrict__ k,
                const float* __restrict__ v, const int* __restrict__ mask,
                float* __restrict__ out, float* __restrict__ Opart,
                float* __restrict__ mpart, float* __restrict__ lpart) {
  __shared__ __align__(16) float Ks[2][KT * D_];
  __shared__ __align__(16) float Vs[2][KT * VSTRIDE];

  const int tid  = threadIdx.x;
  const int lane = tid & 31;
  const int wave = tid >> 5;
  const int hi   = lane >> 4;
  const int ln15 = lane & 15;

  const int bs   = blockIdx.x;
  const int sp   = PARTIAL ? (bs % SPLIT) : 0;
  const int tmp  = PARTIAL ? (bs / SPLIT) : bs;
  const int qblk = tmp % NQB;
  const int b    = tmp / NQB;

  constexpr int KSPAN = PARTIAL ? (SK_ / SPLIT) : SK_;
  constexpr int NTt   = KSPAN / KT;         // tiles of keys this block walks
  const int kbase = sp * KSPAN;

  const int qbase = qblk * QPB + wave * QPW;
  const int qrow  = qbase + ln15;

  const float* qg = q + ((size_t)b * SQ_ + qrow) * D_;
  const float* kg = k + ((size_t)b * SK_ + kbase) * D_;
  const float* vg = v + ((size_t)b * SK_ + kbase) * DV_;
  const int*   mg = mask + (size_t)b * SK_ + kbase;

  // Q row resident in VGPRs (lanes L and L+16 duplicate a row)
  float qr[D_];
#pragma unroll
  for (int d = 0; d < D_; d += 4) {
    float4 t = *(const float4*)(qg + d);
    qr[d] = t.x; qr[d + 1] = t.y; qr[d + 2] = t.z; qr[d + 3] = t.w;
  }

  v8f acc[4];
#pragma unroll
  for (int n = 0; n < 4; ++n) {
    v8f z = {0.f, 0.f, 0.f, 0.f, 0.f, 0.f, 0.f, 0.f};
    acc[n] = z;
  }
  float m_run = -INFINITY;
  float l_run = 0.0f;

  // async tile fill: 1024 b128 chunks / 64 threads = 16 per thread per tile
  auto issue_tile = [&](int buf, int k0) {
#pragma unroll
    for (int i = 0; i < 8; ++i) {                 // K tile: 512 chunks
      int chunk = tid + i * BLOCK;
      int key = chunk >> 4, c4 = chunk & 15;
      uint32_t loff = (uint32_t)(uintptr_t)&Ks[buf][key * D_ + c4 * 4];
      async_ld_b128(loff, (uint64_t)(kg + (size_t)(k0 + key) * D_ + c4 * 4));
    }
#pragma unroll
    for (int i = 0; i < 8; ++i) {                 // V tile: 512 chunks
      int chunk = tid + i * BLOCK;
      int key = chunk >> 4, c4 = chunk & 15;
      uint32_t loff = (uint32_t)(uintptr_t)&Vs[buf][key * VSTRIDE + c4 * 4];
      async_ld_b128(loff, (uint64_t)(vg + (size_t)(k0 + key) * DV_ + c4 * 4));
    }
  };

  issue_tile(0, 0);

  for (int t = 0; t < NTt; ++t) {
    const int buf = t & 1;
    if (t + 1 < NTt) {
      issue_tile(buf ^ 1, (t + 1) * KT);
      asm volatile("s_wait_asynccnt 16" ::: "memory");
    } else {
      asm volatile("s_wait_asynccnt 0" ::: "memory");
    }
    __syncthreads();

    const float* Kt = Ks[buf];
    const float* Vt = Vs[buf];

#pragma unroll 1                       // keep one 16-key chunk live at a time
    for (int c = 0; c < KT / 16; ++c) {
      float s[8];
      int   mk[8];
#pragma unroll
      for (int j = 0; j < 8; ++j) {
        const int sel = (j & 1) ? (hi ? 3 : 1) : (hi ? 2 : 0);
        const int kl  = c * 16 + 4 * (j >> 1) + sel;
        const float4* kr4 = (const float4*)(Kt + kl * D_);
        // 4 independent partial sums -> short dep chains, low liveness
        float d0 = 0.f, d1 = 0.f, d2 = 0.f, d3 = 0.f;
#pragma unroll
        for (int d4 = 0; d4 < D_ / 4; ++d4) {
          float4 kv = kr4[d4];
          d0 += fabsf(qr[4 * d4 + 0] - kv.x);
          d1 += fabsf(qr[4 * d4 + 1] - kv.y);
          d2 += fabsf(qr[4 * d4 + 2] - kv.z);
          d3 += fabsf(qr[4 * d4 + 3] - kv.w);
        }
        const float dist = (d0 + d1) + (d2 + d3);
        mk[j] = mg[t * KT + kl];
        s[j]  = mk[j] ? -dist : NEG_;
      }

      // online softmax
      float mloc = s[0];
#pragma unroll
      for (int j = 1; j < 8; ++j) mloc = fmaxf(mloc, s[j]);
      mloc = fmaxf(mloc, __shfl_xor(mloc, 16, 32));
      const float m_new = fmaxf(m_run, mloc);
      const float scale = __expf(m_run - m_new);

      float p[8];
      float lsum = 0.f;
#pragma unroll
      for (int j = 0; j < 8; ++j) {
        p[j] = mk[j] ? __expf(s[j] - m_new) : 0.f;
        lsum += p[j];
      }
      lsum += __shfl_xor(lsum, 16, 32);
      l_run = l_run * scale + lsum;
      m_run = m_new;

      float scr[8];
#pragma unroll
      for (int r = 0; r < 8; ++r) scr[r] = __shfl(scale, hi ? (r + 8) : r, 32);
#pragma unroll
      for (int n = 0; n < 4; ++n)
#pragma unroll
        for (int r = 0; r < 8; ++r) acc[n][r] *= scr[r];

      // P(16x16) x V(16x64): 4 K-steps x 4 dv tiles of v_wmma_f32_16x16x4_f32
#pragma unroll
      for (int j = 0; j < 4; ++j) {
        const int r0 = c * 16 + 4 * j + (hi ? 2 : 0);
        const int r1 = c * 16 + 4 * j + (hi ? 3 : 1);
        v2f a;
        a.x = p[2 * j];
        a.y = p[2 * j + 1];
#pragma unroll
        for (int n = 0; n < 4; ++n) {
          v2f bv;
          bv.x = Vt[r0 * VSTRIDE + n * 16 + ln15];
          bv.y = Vt[r1 * VSTRIDE + n * 16 + ln15];
          acc[n] = __builtin_amdgcn_wmma_f32_16x16x4_f32(
              false, a, false, bv, (short)0, acc[n], false, false);
        }
      }
    }
    __syncthreads();
  }

  if constexpr (PARTIAL) {
    const int tile16 = qblk * (QPB / 16) + wave;           // 16-row tile id in batch
    const int pidx   = (b * NT16 + tile16) * SPLIT + sp;
    float* Op = Opart + (size_t)pidx * 16 * DV_;
#pragma unroll
    for (int r = 0; r < 8; ++r)
#pragma unroll
      for (int n = 0; n < 4; ++n)
        Op[(size_t)(r + hi * 8) * DV_ + n * 16 + ln15] = acc[n][r];
    if (!hi) {                       // one copy per query row
      mpart[pidx * 16 + ln15] = m_run;
      lpart[pidx * 16 + ln15] = l_run;
    }
  } else {
    const float linv = 1.0f / l_run;
    float li[8];
#pragma unroll
    for (int r = 0; r < 8; ++r) li[r] = __shfl(linv, hi ? (r + 8) : r, 32);
    float* og = out + (size_t)b * SQ_ * DV_;
#pragma unroll
    for (int r = 0; r < 8; ++r) {
      const int row = qbase + r + hi * 8;
#pragma unroll
      for (int n = 0; n < 4; ++n)
        og[(size_t)row * DV_ + n * 16 + ln15] = acc[n][r] * li[r];
    }
  }
}

// merge SPLIT partials per query row: out = sum_s e^{m_s-M} O_s / sum_s e^{m_s-M} l_s
__global__ void mattn_combine(const float* __restrict__ Opart,
                              const float* __restrict__ mpart,
                              const float* __restrict__ lpart,
                              float* __restrict__ out) {
  const int idx = blockIdx.x * blockDim.x + threadIdx.x;
  if (idx >= B_ * SQ_ * DV_) return;
  const int col = idx % DV_;
  const int row = idx / DV_;                 // b*SQ + q
  const int b = row / SQ_, qq = row % SQ_;
  const int tile = qq >> 4, r = qq & 15;
  const int p0 = (b * NT16 + tile) * SPLIT;

  float M = -INFINITY;
#pragma unroll
  for (int s = 0; s < SPLIT; ++s) M = fmaxf(M, mpart[(p0 + s) * 16 + r]);
  float L = 0.f, val = 0.f;
#pragma unroll
  for (int s = 0; s < SPLIT; ++s) {
    const float w = __expf(mpart[(p0 + s) * 16 + r] - M);
    L += w * lpart[(p0 + s) * 16 + r];
    val += w * Opart[((size_t)(p0 + s) * 16 + r) * DV_ + col];
  }
  out[idx] = val / L;
}

extern "C" void kernel_launch(void* const* d_in, const int* in_sizes, int n_in,
                              void* d_out, int out_size, void* d_ws, size_t ws_size,
                              hipStream_t stream) {
  const float* q    = (const float*)d_in[0];
  const float* k    = (const float*)d_in[1];
  const float* v    = (const float*)d_in[2];
  const int*   mask = (const int*)d_in[3];
  float* out = (float*)d_out;

  const size_t nPart  = (size_t)B_ * NT16 * SPLIT;           // 512 partial tiles
  const size_t needO  = nPart * 16 * DV_ * sizeof(float);    // ~2 MB
  const size_t needML = nPart * 16 * sizeof(float);          // 32 KB each

  if (ws_size >= needO + 2 * needML) {
    float* Opart = (float*)d_ws;
    float* mpart = (float*)((char*)d_ws + needO);
    float* lpart = mpart + nPart * 16;
    hipLaunchKernelGGL((mattn_main<true>), dim3(B_ * NQB * SPLIT), dim3(BLOCK),
                       0, stream, q, k, v, mask, out, Opart, mpart, lpart);
    const int tot = B_ * SQ_ * DV_;
    hipLaunchKernelGGL(mattn_combine, dim3((tot + 255) / 256), dim3(256),
                       0, stream, Opart, mpart, lpart, out);
  } else {
    hipLaunchKernelGGL((mattn_main<false>), dim3(B_ * NQB), dim3(BLOCK),
                       0, stream, q, k, v, mask, out,
                       (float*)nullptr, (float*)nullptr, (float*)nullptr);
  }
}